// Factorized_Linear_12678743457792
// MI455X (gfx1250) — compile-verified
//
#include <hip/hip_runtime.h>

typedef __attribute__((ext_vector_type(2))) float v2f;
typedef __attribute__((ext_vector_type(8))) float v8f;

#if defined(__gfx1250__) && !__has_builtin(__builtin_amdgcn_wmma_f32_16x16x4_f32)
#error "missing __builtin_amdgcn_wmma_f32_16x16x4_f32 on gfx1250"
#endif

namespace {
constexpr int Bsz = 128;   // batch  (GEMM M)
constexpr int Isz = 64;    // kept in-axis (one GEMM per i)
constexpr int Ksz = 1024;  // reduction
constexpr int Osz = 1024;  // out-axis (GEMM N)

constexpr int TN     = 128;        // N tile per workgroup
constexpr int KC     = 32;         // K chunk staged in LDS
constexpr int KCP    = KC + 4;     // padded LDS row stride (dwords), conflict-free frags
constexpr int NCHUNK = Ksz / KC;   // 32

constexpr int ATILE = Bsz * KCP;   // floats per A buffer
constexpr int BTILE = TN  * KCP;   // floats per B buffer
constexpr unsigned SMEM_BYTES = (unsigned)(2 * ATILE + 2 * BTILE) * sizeof(float); // 72 KB
}

template <int N>
__device__ __forceinline__ void wait_async() {
#if __has_builtin(__builtin_amdgcn_s_wait_asynccnt)
    __builtin_amdgcn_s_wait_asynccnt(N);
#else
    asm volatile("s_wait_asynccnt %0" :: "i"(N) : "memory");
#endif
}

// Stage this thread's 64B half-row of A and of B into LDS, asynchronously.
// INST_OFFSET adds to both the LDS and the global address (ISA 10.7/15.18),
// so one address pair + offset:0/16/32/48 covers the 64 bytes.
// Weight (B) is streamed exactly once -> non-temporal hint to protect L2.
__device__ __forceinline__ void async_stage(uint32_t ldsA, uint32_t ldsB,
                                            unsigned long long gA, unsigned long long gB) {
    asm volatile(
        "global_load_async_to_lds_b128 %0, %2, off\n\t"
        "global_load_async_to_lds_b128 %0, %2, off offset:16\n\t"
        "global_load_async_to_lds_b128 %0, %2, off offset:32\n\t"
        "global_load_async_to_lds_b128 %0, %2, off offset:48\n\t"
        "global_load_async_to_lds_b128 %1, %3, off th:TH_LOAD_NT\n\t"
        "global_load_async_to_lds_b128 %1, %3, off offset:16 th:TH_LOAD_NT\n\t"
        "global_load_async_to_lds_b128 %1, %3, off offset:32 th:TH_LOAD_NT\n\t"
        "global_load_async_to_lds_b128 %1, %3, off offset:48 th:TH_LOAD_NT"
        :: "v"(ldsA), "v"(ldsB), "v"(gA), "v"(gB)
        : "memory");
}

__global__ __launch_bounds__(256)
void factorized_linear_wmma(const float* __restrict__ xg,
                            const float* __restrict__ wg,
                            const float* __restrict__ bg,
                            float* __restrict__ outg)
{
    extern __shared__ __align__(16) float smem[];
    float* lA0 = smem;                       // x tile, buffer 0
    float* lA1 = smem + ATILE;               // x tile, buffer 1
    float* lB0 = smem + 2 * ATILE;           // weight tile, buffer 0
    float* lB1 = smem + 2 * ATILE + BTILE;   // weight tile, buffer 1

    const int i    = blockIdx.y;            // 0..63
    const int nb   = blockIdx.x * TN;       // base output channel
    const int tid  = threadIdx.x;           // 0..255
    const int wave = tid >> 5;              // 0..7  -> 16-wide N column
    const int lane = tid & 31;
    const int l16  = lane & 15;
    const int hi   = lane >> 4;             // 0/1

    // Loader mapping: thread t stages row r = t>>1, 16-float half h = t&1.
    const int lr = tid >> 1;                // 0..127
    const int lh = (tid & 1) * 16;          // 0 or 16

    const float* gA = xg + ((size_t)lr * Isz + i) * Ksz + lh;
    const float* gB = wg + ((size_t)(nb + lr) * Isz + i) * Ksz + lh;

    // LDS byte addresses for the async destination (low 32 bits of generic ptr).
    const uint32_t sa0 = (uint32_t)(uintptr_t)(lA0 + lr * KCP + lh);
    const uint32_t sa1 = (uint32_t)(uintptr_t)(lA1 + lr * KCP + lh);
    const uint32_t sb0 = (uint32_t)(uintptr_t)(lB0 + lr * KCP + lh);
    const uint32_t sb1 = (uint32_t)(uintptr_t)(lB1 + lr * KCP + lh);

    // Accumulators preloaded with bias (broadcast over batch).
    const int o  = nb + wave * 16 + l16;
    const float bv = bg[(size_t)o * Isz + i];
    v8f acc[8];
#pragma unroll
    for (int mt = 0; mt < 8; ++mt)
#pragma unroll
        for (int v = 0; v < 8; ++v) acc[mt][v] = bv;

    // Prologue: chunk 0 -> buffer 0 (8 async ops per wave).
    async_stage(sa0, sb0, (unsigned long long)gA, (unsigned long long)gB);

    for (int c = 0; c < NCHUNK; ++c) {
        // Issue chunk c+1 into the other buffer, then wait for chunk c only
        // (async completions are in-order per wave: <=8 outstanding == c done).
        if (c + 1 < NCHUNK) {
            const int kn = (c + 1) * KC;
            if ((c + 1) & 1)
                async_stage(sa1, sb1, (unsigned long long)(gA + kn),
                                       (unsigned long long)(gB + kn));
            else
                async_stage(sa0, sb0, (unsigned long long)(gA + kn),
                                       (unsigned long long)(gB + kn));
            wait_async<8>();
        } else {
            wait_async<0>();
        }
        __syncthreads();   // all waves' chunk-c data resident in LDS

        const float* la = (c & 1) ? lA1 : lA0;
        const float* lb = (c & 1) ? lB1 : lB0;

        // 8 k-steps x 8 M-subtiles of V_WMMA_F32_16X16X4_F32
        // A frag (16x4 f32): lane holds A[m = l16][k = kk + 2*hi + {0,1}]
        // B frag (4x16 f32): lane holds B[k = kk + 2*hi + {0,1}][n = l16]
#pragma unroll
        for (int kk = 0; kk < KC; kk += 4) {
            const v2f bf = *(const v2f*)&lb[(wave * 16 + l16) * KCP + kk + 2 * hi];
#pragma unroll
            for (int mt = 0; mt < 8; ++mt) {
                const v2f af = *(const v2f*)&la[(mt * 16 + l16) * KCP + kk + 2 * hi];
                acc[mt] = __builtin_amdgcn_wmma_f32_16x16x4_f32(
                    false, af, false, bf, (short)0, acc[mt], false, false);
            }
        }

        __syncthreads();   // chunk-c buffer free for reuse at iteration c+1
    }

    // Epilogue: D layout -> out[b][o][i]; lane: n = l16, vgpr v: m = v + 8*hi.
#pragma unroll
    for (int mt = 0; mt < 8; ++mt) {
#pragma unroll
        for (int v = 0; v < 8; ++v) {
            const int b = mt * 16 + hi * 8 + v;
            outg[((size_t)b * Osz + o) * Isz + i] = acc[mt][v];
        }
    }
}

extern "C" void kernel_launch(void* const* d_in, const int* in_sizes, int n_in,
                              void* d_out, int out_size, void* d_ws, size_t ws_size,
                              hipStream_t stream) {
    const float* x = (const float*)d_in[0];   // [B, I, K] f32
    const float* w = (const float*)d_in[1];   // [O, I, K] f32
    const float* b = (const float*)d_in[2];   // [O, I]    f32
    float* out     = (float*)d_out;           // [B, O, I] f32

    dim3 grid(Osz / TN, Isz);   // (8, 64)
    dim3 block(256);
    factorized_linear_wmma<<<grid, block, SMEM_BYTES, stream>>>(x, w, b, out);
}